// ChamferDistance_56616258895919
// MI455X (gfx1250) — compile-verified
//
#include <hip/hip_runtime.h>

typedef float v2f __attribute__((ext_vector_type(2)));
typedef float v8f __attribute__((ext_vector_type(8)));

#define WAVES_PER_BLOCK 8
#define TILE 16

// Pack each point into (x, y, z, ||p||^2), 16 bytes, so the WMMA hot loop
// needs exactly one global_load_b64 per lane per 16x16 tile.
__global__ void chamfer_prep(const float* __restrict__ xyz,
                             float* __restrict__ aug, int npts) {
  int i = blockIdx.x * blockDim.x + threadIdx.x;
  if (i < npts) {
    float x = xyz[3 * i + 0];
    float y = xyz[3 * i + 1];
    float z = xyz[3 * i + 2];
    aug[4 * i + 0] = x;
    aug[4 * i + 1] = y;
    aug[4 * i + 2] = z;
    aug[4 * i + 3] = x * x + y * y + z * z;
  }
}

// One wave owns 16 query points (the N/column dim of the WMMA tile) and
// streams all targets (the M/row dim, 16 per iteration) through
// v_wmma_f32_16x16x4_f32:  D[m,n] = ||t_m||^2 - 2 * t_m . q_n
// (augmented K=4 trick; +||q_n||^2 added once at the end).
// Column mins are in-lane: 8 v_min per tile, one shfl_xor(16) at the end.
__global__ void chamfer_min_pass(const float* __restrict__ augT,
                                 const float* __restrict__ augQ,
                                 int numT, int numQ,
                                 float* __restrict__ out, float invQ) {
  const int lane = threadIdx.x & 31;
  const int wave = threadIdx.x >> 5;
  const int half = lane >> 4;   // 0: K=0,1 lanes  1: K=2,3 lanes
  const int sub  = lane & 15;
  const int b    = blockIdx.y;

  const float* __restrict__ augQb = augQ + (size_t)b * numQ * 4;
  const float* __restrict__ augTb = augT + (size_t)b * numT * 4;

  const int qBase = (blockIdx.x * WAVES_PER_BLOCK + wave) * TILE;
  int qIdx = qBase + sub;
  qIdx = qIdx < numQ ? qIdx : (numQ - 1);   // branchless clamp, keep EXEC full

  // B matrix regs: lo half = (-2x, -2y), hi half = (-2z, 1.0)
  v2f qc = *(const v2f*)(augQb + 4 * (size_t)qIdx + 2 * half);
  v2f bmat;
  bmat[0] = -2.0f * qc[0];
  bmat[1] = half ? 1.0f : (-2.0f * qc[1]);

  float acc = 3.0e38f;
  for (int t0 = 0; t0 < numT; t0 += TILE) {
    int tIdx = t0 + sub;
    // A matrix regs: lo half = (x, y), hi half = (z, ||t||^2)
    v2f amat = *(const v2f*)(augTb + 4 * (size_t)tIdx + 2 * half);
    v8f c = {};
    c = __builtin_amdgcn_wmma_f32_16x16x4_f32(
        /*neg_a=*/false, amat, /*neg_b=*/false, bmat,
        /*c_mod=*/(short)0, c, /*reuse_a=*/false, /*reuse_b=*/false);
    float m01 = fminf(c[0], c[1]);
    float m23 = fminf(c[2], c[3]);
    float m45 = fminf(c[4], c[5]);
    float m67 = fminf(c[6], c[7]);
    acc = fminf(acc, fminf(fminf(m01, m23), fminf(m45, m67)));
  }
  // combine row halves M=0..7 (lanes 0-15) with M=8..15 (lanes 16-31)
  acc = fminf(acc, __shfl_xor(acc, 16, 32));

  float val = 0.0f;
  if (lane < 16 && (qBase + sub) < numQ) {
    val = (acc + augQb[4 * (size_t)qIdx + 3]) * invQ;  // + ||q||^2, scaled for mean
  }
  // wave-level sum (hi lanes contribute 0)
  for (int off = 1; off < 32; off <<= 1) val += __shfl_xor(val, off, 32);

  __shared__ float wsum[WAVES_PER_BLOCK];
  if (lane == 0) wsum[wave] = val;
  __syncthreads();
  if (threadIdx.x == 0) {
    float s = 0.0f;
    for (int w = 0; w < WAVES_PER_BLOCK; ++w) s += wsum[w];
    atomicAdd(out + b, s);
  }
}

extern "C" void kernel_launch(void* const* d_in, const int* in_sizes, int n_in,
                              void* d_out, int out_size, void* d_ws, size_t ws_size,
                              hipStream_t stream) {
  const float* xyz1 = (const float*)d_in[0];
  const float* xyz2 = (const float*)d_in[1];
  float* out = (float*)d_out;

  const int B = out_size;                 // 4
  const int N = in_sizes[0] / (3 * B);    // 8192
  const int M = in_sizes[1] / (3 * B);    // 8192

  float* aug1 = (float*)d_ws;                       // B*N*4 floats
  float* aug2 = aug1 + (size_t)B * N * 4;           // B*M*4 floats

  hipMemsetAsync(d_out, 0, (size_t)out_size * sizeof(float), stream);

  const int total1 = B * N, total2 = B * M;
  chamfer_prep<<<(total1 + 255) / 256, 256, 0, stream>>>(xyz1, aug1, total1);
  chamfer_prep<<<(total2 + 255) / 256, 256, 0, stream>>>(xyz2, aug2, total2);

  const int qPerBlock = WAVES_PER_BLOCK * TILE;     // 128 queries per block

  // term 1: for each xyz1 point, min over xyz2; mean over N
  dim3 g1((N + qPerBlock - 1) / qPerBlock, B);
  chamfer_min_pass<<<g1, WAVES_PER_BLOCK * 32, 0, stream>>>(
      aug2, aug1, M, N, out, 1.0f / (float)N);

  // term 2: for each xyz2 point, min over xyz1; mean over M
  dim3 g2((M + qPerBlock - 1) / qPerBlock, B);
  chamfer_min_pass<<<g2, WAVES_PER_BLOCK * 32, 0, stream>>>(
      aug1, aug2, N, M, out, 1.0f / (float)M);
}